// Blur_80281528697499
// MI455X (gfx1250) — compile-verified
//
#include <hip/hip_runtime.h>
#include <hip/hip_bf16.h>

// Blur / UpFirDn2D (up=2, dn=2, k=[1,3,3,1] outer) collapses to a 2x2 stencil:
//   out[m,n] = f11*x[m,n] + f13*x[m,n+1] + f31*x[m+1,n] + f33*x[m+1,n+1]
// with f11=filt[c,0,1,1], f13=filt[c,0,1,3], f31=filt[c,0,3,1], f33=filt[c,0,3,3].
// Bandwidth-bound (AI ~0.5 FLOP/byte, floor ~11.5us at 23.3 TB/s): stream each
// 128x128 plane through LDS once via CDNA5 async global->LDS loads (no VGPR
// staging), compute from LDS, store b128.

#define BB  8
#define CC  256
#define HH  128
#define WW  128
#define PLANE_ELEMS (HH * WW)          // 16384 floats = 64 KB
#define THREADS 256                     // 8 waves (wave32)

// b128 async copy operates on int-vector-of-4 payloads per the builtin's
// signature: (addrspace(1) v4i* gsrc, addrspace(3) v4i* ldst, imm offset, imm cpol)
typedef int v4i __attribute__((vector_size(16)));
typedef __attribute__((address_space(1))) v4i* g_v4i_ptr;
typedef __attribute__((address_space(3))) v4i* l_v4i_ptr;

__global__ __launch_bounds__(THREADS)
void blur_updown_kernel(const float* __restrict__ x,
                        const float* __restrict__ filt,
                        float* __restrict__ out) {
    __shared__ float tile[PLANE_ELEMS];  // 64 KB: one full (H,W) plane

    const unsigned tid   = threadIdx.x;
    const unsigned plane = blockIdx.x;               // plane = b*C + c, 0..2047
    const size_t   base  = (size_t)plane * PLANE_ELEMS;

    // ---- Phase 1: async DMA plane -> LDS (contiguous, 16B per lane per op) ----
    // Low 32 bits of the flat shared-aperture address == LDS byte offset.
    const unsigned lds0  = (unsigned)(size_t)(void*)tile;
    const unsigned long long gbase =
        (unsigned long long)(size_t)(const void*)(x + base);

#pragma unroll
    for (int it = 0; it < PLANE_ELEMS / 4 / THREADS; ++it) {   // 16 iterations
        const unsigned f4  = (unsigned)it * THREADS + tid;     // float4 index
        const unsigned lds = lds0 + f4 * 16u;
        const unsigned long long g = gbase + (unsigned long long)f4 * 16ull;
#if __has_builtin(__builtin_amdgcn_global_load_async_to_lds_b128)
        __builtin_amdgcn_global_load_async_to_lds_b128(
            (g_v4i_ptr)(size_t)g, (l_v4i_ptr)(size_t)lds, 0, 0);
#else
        asm volatile("global_load_async_to_lds_b128 %0, %1, off"
                     :: "v"(lds), "v"(g) : "memory");
#endif
    }
#if __has_builtin(__builtin_amdgcn_s_wait_asynccnt)
    __builtin_amdgcn_s_wait_asynccnt(0);
#else
    asm volatile("s_wait_asynccnt 0" ::: "memory");
#endif
    __syncthreads();

    // ---- Phase 2: 2x2 stencil from LDS ----
    const int w = (int)(tid >> 5);          // wave id 0..7
    const int l = (int)(tid & 31);          // lane 0..31 -> columns 4l..4l+3
    const int c = (int)(plane % CC);

    // Block-uniform taps -> scalar loads.
    const float f11 = filt[c * 16 + 5];
    const float f13 = filt[c * 16 + 7];
    const float f31 = filt[c * 16 + 13];
    const float f33 = filt[c * 16 + 15];

    const float4* __restrict__ t4 = (const float4*)tile;

#pragma unroll 4
    for (int k = 0; k < HH / 8; ++k) {      // 16 rows per wave
        const int m = w + k * 8;

        const float4 a  = t4[m * (WW / 4) + l];                        // x[m, 4l..4l+3]
        const float  a4 = (l < 31) ? tile[m * WW + 4 * l + 4] : 0.0f;  // x[m, 4l+4]

        float4 b;
        float  b4;
        if (m < HH - 1) {
            b  = t4[(m + 1) * (WW / 4) + l];
            b4 = (l < 31) ? tile[(m + 1) * WW + 4 * l + 4] : 0.0f;
        } else {
            b  = make_float4(0.0f, 0.0f, 0.0f, 0.0f);
            b4 = 0.0f;
        }

        float4 o;
        o.x = f11 * a.x + f13 * a.y + f31 * b.x + f33 * b.y;
        o.y = f11 * a.y + f13 * a.z + f31 * b.y + f33 * b.z;
        o.z = f11 * a.z + f13 * a.w + f31 * b.z + f33 * b.w;
        o.w = f11 * a.w + f13 * a4  + f31 * b.w + f33 * b4;

        *(float4*)(out + base + (size_t)m * WW + 4 * l) = o;
    }
}

extern "C" void kernel_launch(void* const* d_in, const int* in_sizes, int n_in,
                              void* d_out, int out_size, void* d_ws, size_t ws_size,
                              hipStream_t stream) {
    (void)in_sizes; (void)n_in; (void)out_size; (void)d_ws; (void)ws_size;
    const float* x    = (const float*)d_in[0];
    const float* filt = (const float*)d_in[1];
    float*       out  = (float*)d_out;

    dim3 grid(BB * CC);     // 2048 planes
    dim3 block(THREADS);    // 256 threads = 8 wave32
    blur_updown_kernel<<<grid, block, 0, stream>>>(x, filt, out);
}